// HybridLightGCN_5600637354192
// MI455X (gfx1250) — compile-verified
//
#include <hip/hip_runtime.h>

typedef __attribute__((ext_vector_type(2))) float v2f;
typedef __attribute__((ext_vector_type(8))) float v8f;

constexpr int NUM_USERS = 100000;
constexpr int NUM_ITEMS = 50000;
constexpr int N_NODES   = NUM_USERS + NUM_ITEMS;   // 150000
constexpr int DIM       = 64;
constexpr int SBERT     = 384;
constexpr int NEDGE     = 2000000;
constexpr int NLAYERS   = 3;

// ---------------------------------------------------------------------------
// Utility: zero a float buffer (d_ws is poisoned, so we must clear ourselves)
// ---------------------------------------------------------------------------
__global__ void zero_f(float* __restrict__ p, int n) {
  int i = blockIdx.x * blockDim.x + threadIdx.x;
  if (i < n) p[i] = 0.0f;
}

// ---------------------------------------------------------------------------
// emb0[0:U] = user_emb ; acc[0:U] = user_emb
// ---------------------------------------------------------------------------
__global__ void init_user(const float* __restrict__ u,
                          float* __restrict__ emb0, float* __restrict__ acc) {
  int i = blockIdx.x * blockDim.x + threadIdx.x;
  if (i < NUM_USERS * DIM) {
    float v = u[i];
    emb0[i] = v;
    acc[i]  = v;
  }
}

// ---------------------------------------------------------------------------
// book_emb = book_bert @ w_proj.T  via V_WMMA_F32_16X16X4_F32
// One wave per 16x16 output tile; 4 waves per block cover the 4 N-tiles of
// one M-tile row (shares A rows in L1).  K-loop: 384/4 = 96 chained WMMAs.
//
// A (16x4 f32) per-lane layout: lanes 0-15 -> K = k+0,k+1 ; lanes 16-31 ->
// K = k+2,k+3 (M = lane&15).  B (4x16) mirrored: lane r of half h supplies
// B[k+2h + {0,1}, n0+r] = w_proj[n0+r, k+2h + {0,1}] -> contiguous 8B loads.
// C/D (16x16 f32): vgpr i, half h -> row i+8h, col = lane&15.
// ---------------------------------------------------------------------------
__global__ __launch_bounds__(128)
void wmma_proj(const float* __restrict__ bert, const float* __restrict__ w,
               float* __restrict__ emb0, float* __restrict__ acc) {
  const int lane = threadIdx.x & 31;
  const int wave = threadIdx.x >> 5;          // 0..3 -> N tile
  const int m0   = blockIdx.x * 16;           // item-row tile
  const int n0   = wave * 16;
  const int half = lane >> 4;
  const int r    = lane & 15;

  const float* arow = bert + (size_t)(m0 + r) * SBERT + 2 * half;
  const float* brow = w    + (size_t)(n0 + r) * SBERT + 2 * half;

  v8f c = {};
#pragma unroll 8
  for (int k = 0; k < SBERT; k += 4) {
    v2f a = *(const v2f*)(arow + k);
    v2f b = *(const v2f*)(brow + k);
    c = __builtin_amdgcn_wmma_f32_16x16x4_f32(
        /*neg_a=*/false, a, /*neg_b=*/false, b,
        /*c_mod=*/(short)0, c, /*reuse_a=*/false, /*reuse_b=*/false);
  }

  // row of output element i: NUM_USERS + m0 + i + 8*half ; col: n0 + r
  const size_t base = (size_t)(NUM_USERS + m0 + 8 * half) * DIM + (size_t)(n0 + r);
#pragma unroll
  for (int i = 0; i < 8; ++i) {
    float v = c[i];
    emb0[base + (size_t)i * DIM] = v;
    acc [base + (size_t)i * DIM] = v;
  }
}

// ---------------------------------------------------------------------------
// deg[dst[e]] += 1  (symmetric gcn_norm, degrees over dst like the reference)
// ---------------------------------------------------------------------------
__global__ void deg_count(const int* __restrict__ dst, float* __restrict__ deg) {
  int e = blockIdx.x * blockDim.x + threadIdx.x;
  if (e < NEDGE) atomicAdd(&deg[dst[e]], 1.0f);
}

// deg -> dis = deg>0 ? rsqrt(max(deg,1)) : 0  (in place)
__global__ void deg_to_dis(float* __restrict__ deg) {
  int i = blockIdx.x * blockDim.x + threadIdx.x;
  if (i < N_NODES) {
    float d = deg[i];
    deg[i] = (d > 0.0f) ? rsqrtf(fmaxf(d, 1.0f)) : 0.0f;
  }
}

// norm[e] = dis[src[e]] * dis[dst[e]]
__global__ void edge_norm(const int* __restrict__ src, const int* __restrict__ dst,
                          const float* __restrict__ dis, float* __restrict__ norm) {
  int e = blockIdx.x * blockDim.x + threadIdx.x;
  if (e < NEDGE) norm[e] = dis[src[e]] * dis[dst[e]];
}

// ---------------------------------------------------------------------------
// xn[dst[e]] += norm[e] * x[src[e]]   (16 lanes per edge, float4 per lane)
// x / xn are 38.4 MB each -> L2-resident on MI455X (192 MB L2), so the random
// gathers and the f32 atomic adds stay on-chip; HBM sees only the edge stream.
// ---------------------------------------------------------------------------
__global__ void spmm_scatter(const int* __restrict__ src, const int* __restrict__ dst,
                             const float* __restrict__ norm,
                             const float* __restrict__ x, float* __restrict__ xn) {
  int t = blockIdx.x * blockDim.x + threadIdx.x;
  int e = t >> 4;
  if (e >= NEDGE) return;
  int c = (t & 15) << 2;
  float wgt = norm[e];
  float4 v = *(const float4*)(x + (size_t)src[e] * DIM + c);
  float* o = xn + (size_t)dst[e] * DIM + c;
  atomicAdd(o + 0, wgt * v.x);
  atomicAdd(o + 1, wgt * v.y);
  atomicAdd(o + 2, wgt * v.z);
  atomicAdd(o + 3, wgt * v.w);
}

__global__ void acc_add(float* __restrict__ acc, const float* __restrict__ xn) {
  int i = blockIdx.x * blockDim.x + threadIdx.x;
  if (i < N_NODES * DIM) acc[i] += xn[i];
}

__global__ void scale_k(float* __restrict__ acc) {
  int i = blockIdx.x * blockDim.x + threadIdx.x;
  if (i < N_NODES * DIM) acc[i] *= 0.25f;  // 1/(NUM_LAYERS+1)
}

// ---------------------------------------------------------------------------
extern "C" void kernel_launch(void* const* d_in, const int* in_sizes, int n_in,
                              void* d_out, int out_size, void* d_ws, size_t ws_size,
                              hipStream_t stream) {
  const int*   ei    = (const int*)d_in[0];      // (2, E) int32
  const int*   src   = ei;
  const int*   dst   = ei + NEDGE;
  const float* user  = (const float*)d_in[1];    // (100000, 64)
  const float* bert  = (const float*)d_in[2];    // (50000, 384)
  const float* wproj = (const float*)d_in[3];    // (64, 384)

  float* emb0 = (float*)d_out;                         // first output
  float* acc  = emb0 + (size_t)N_NODES * DIM;          // second output (acc in place)

  // workspace carve-up (floats), 256B-aligned chunks
  float* deg  = (float*)d_ws;                          // N_NODES
  float* norm = deg  + ((N_NODES + 63) & ~63);         // NEDGE
  float* xA   = norm + ((NEDGE + 63) & ~63);           // N_NODES*DIM
  float* xB   = xA   + (size_t)N_NODES * DIM;          // N_NODES*DIM

  const int B = 256;
  const int NV = N_NODES * DIM;

  zero_f   <<<(N_NODES + B - 1) / B, B, 0, stream>>>(deg, N_NODES);
  init_user<<<(NUM_USERS * DIM + B - 1) / B, B, 0, stream>>>(user, emb0, acc);
  wmma_proj<<<NUM_ITEMS / 16, 128, 0, stream>>>(bert, wproj, emb0, acc);

  deg_count <<<(NEDGE + B - 1) / B, B, 0, stream>>>(dst, deg);
  deg_to_dis<<<(N_NODES + B - 1) / B, B, 0, stream>>>(deg);
  edge_norm <<<(NEDGE + B - 1) / B, B, 0, stream>>>(src, dst, deg, norm);

  const float* xcur = emb0;
  float* bufs[2] = {xA, xB};
  for (int l = 0; l < NLAYERS; ++l) {
    float* xn = bufs[l & 1];
    zero_f      <<<(NV + B - 1) / B, B, 0, stream>>>(xn, NV);
    spmm_scatter<<<(NEDGE * 16 + B - 1) / B, B, 0, stream>>>(src, dst, norm, xcur, xn);
    acc_add     <<<(NV + B - 1) / B, B, 0, stream>>>(acc, xn);
    xcur = xn;
  }
  scale_k<<<(NV + B - 1) / B, B, 0, stream>>>(acc);
}